// Net_52913997087092
// MI455X (gfx1250) — compile-verified
//
#include <hip/hip_runtime.h>
#include <hip/hip_bf16.h>

// Problem constants (from reference)
#define BSZ  2048
#define DD   256
#define LL   48
#define HH   256
#define OO   10
#define WSTR 303          // D + L - 1 (row stride of W)
#define LEAF 12336        // L * (D + 1)
#define KAUG 2352         // L*L + L (C-flat + offsets)
#define KPAD 2368         // padded to multiple of 64 for aligned WMMA frags

typedef __attribute__((ext_vector_type(2))) float v2f;
typedef __attribute__((ext_vector_type(8))) float v8f;

// ---------------------------------------------------------------------------
// Kernel 1: per-sample sequential recursion.
//   P_i   = x . wx_i
//   z_i   = P_i + sum_{k<i} a_k wh_i[k] + b_i ;  s_i = z_i>0 ; a_i = relu(z_i)
//   C(i,j)= s_i (delta_ij + sum_{k<i} wh_i[k] C(k,j))     (lower-triangular)
//   offs_i= a_i - sum_j C(i,j) P_j
// Emits Cflat row [C(0,0..47), ..., C(47,0..47), offs(0..47), 16 zeros] (KPAD)
// and the relu mask row (as 0.0/1.0 floats) into the output buffer.
// ---------------------------------------------------------------------------
__global__ __launch_bounds__(64)
void seq_kernel(const float* __restrict__ x, const float* __restrict__ W,
                const float* __restrict__ bvec,
                float* __restrict__ Cflat, float* __restrict__ maskOut)
{
    __shared__ float Cs[LL * LL];
    __shared__ float aS[LL];
    __shared__ float pS[LL];
    __shared__ float offS[LL];
    __shared__ float sS[LL];

    const int b   = blockIdx.x;
    const int tid = threadIdx.x;

    for (int idx = tid; idx < LL * LL; idx += 64) Cs[idx] = 0.0f;

    if (tid < LL) {
        const float* xr = x + b * DD;
        const float* wr = W + tid * WSTR;
        float s = 0.0f;
        #pragma unroll 4
        for (int d = 0; d < DD; ++d) s += xr[d] * wr[d];
        pS[tid] = s;
    }
    __syncthreads();

    for (int i = 0; i < LL; ++i) {
        const float* wh = W + i * WSTR + DD;   // wh_i[k] = W[i, D+k]
        // redundant scalar recomputation across threads (broadcast loads)
        float z = pS[i] + bvec[i];
        for (int kk = 0; kk < i; ++kk) z += aS[kk] * wh[kk];
        const float s = (z > 0.0f) ? 1.0f : 0.0f;
        if (tid == 0) { aS[i] = (z > 0.0f) ? z : 0.0f; sS[i] = s; }
        if (tid < i) {
            const int j = tid;
            float acc = 0.0f;
            for (int kk = j; kk < i; ++kk) acc += wh[kk] * Cs[kk * LL + j];
            Cs[i * LL + j] = s * acc;
        } else if (tid == i) {
            Cs[i * LL + i] = s;
        }
        __syncthreads();
    }

    if (tid < LL) {
        float acc = aS[tid];
        for (int j = 0; j <= tid; ++j) acc -= Cs[tid * LL + j] * pS[j];
        offS[tid] = acc;
        maskOut[b * LL + tid] = sS[tid];
    }
    __syncthreads();

    float* crow = Cflat + b * KPAD;
    for (int idx = tid; idx < KPAD; idx += 64) {
        float v;
        if (idx < LL * LL)    v = Cs[idx];
        else if (idx < KAUG)  v = offS[idx - LL * LL];
        else                  v = 0.0f;
        crow[idx] = v;
    }
}

// ---------------------------------------------------------------------------
// Kernel 2: batch-independent head-weight contraction, LDS-staged.
// One block per i (48 blocks). The 257-column Wb0 slice for block i is
// loaded coalesced into dynamic LDS once (Wb0 read exactly once from HBM),
// then all 48 rows T(i,j,:) are computed from LDS. LDS row stride 257
// (== 1 mod 64) -> conflict-free across lanes.
//   T(i,j,h) = sum_d Wx[j,d] * S[h,d]      rows k = i*48+j
//   U(i,h)   = S[h,256]                    rows k = 2304+i
// ---------------------------------------------------------------------------
__global__ __launch_bounds__(256)
void tmat_kernel(const float* __restrict__ W, const float* __restrict__ Wb0,
                 float* __restrict__ Tmat)
{
    extern __shared__ float S[];           // [HH][257] = 263168 bytes
    const int i = blockIdx.x;
    const int t = threadIdx.x;

    const int base = i * 257;
    for (int idx = t; idx < HH * 257; idx += 256) {
        const int h = idx / 257;
        const int c = idx - h * 257;
        S[idx] = Wb0[h * LEAF + base + c];
    }
    __syncthreads();

    const int h = t;
    Tmat[(LL * LL + i) * HH + h] = S[h * 257 + 256];   // offset column U(i,h)

    const float* Srow = S + h * 257;
    for (int j = 0; j < LL; ++j) {
        const float* wr = W + j * WSTR;    // uniform across lanes -> scalar loads
        float s = 0.0f;
        #pragma unroll 4
        for (int d = 0; d < DD; ++d) s += wr[d] * Srow[d];
        Tmat[(i * LL + j) * HH + h] = s;
    }
}

// Zero the KPAD-KAUG padding rows of Tmat.
__global__ __launch_bounds__(256)
void tpad_kernel(float* __restrict__ Tmat)
{
    Tmat[(KAUG + blockIdx.x) * HH + threadIdx.x] = 0.0f;
}

// ---------------------------------------------------------------------------
// Kernel 3: FP32 WMMA GEMM  D = act(A * B(op) + bias)
//   A: M x K row-major (K multiple of 4, rows 8B aligned)
//   BT=0 -> B is K x N row-major (ldb = N)
//   BT=1 -> B is N x K row-major (ldb = K), i.e. multiply by B^T
// One wave -> 16(M) x 32(N) tile via V_WMMA_F32_16X16X4_F32; 4 waves/block
// stacked in M -> 64x32 block tile; grid 256 blocks -> ~1024 waves in flight.
// Column indices are clamped for loads (columns n>=N are dead accumulator
// lanes, never stored) so the loop is branch/divergence-free. For BT=0 the
// two K-rows of each B fragment share ONE 32-bit induction offset applied to
// two loop-invariant scalar bases (B and B+ldb). With unroll 2 the compiler
// rotates pipeline registers (no copy moves) and keeps partial loadcnt waits.
// ---------------------------------------------------------------------------
template<int BT, int RELU>
__global__ __launch_bounds__(128)
void gemm_wmma(const float* __restrict__ A, const float* __restrict__ B,
               const float* __restrict__ bias, float* __restrict__ Dst,
               int N, int K, int ldb, int ldc)
{
    const int lane  = threadIdx.x & 31;
    const int wave  = threadIdx.x >> 5;
    const int mBase = blockIdx.x * 64 + wave * 16;
    const int nBase = blockIdx.y * 32;

    const int l15   = lane & 15;
    const int kHi   = (lane >> 4) * 2;      // K sub-offset 0 or 2 (A & B frags)
    const int mRow  = mBase + l15;
    const int nCol0 = nBase + l15;
    const int nCol1 = nCol0 + 16;
    const int nc0   = nCol0 < N ? nCol0 : N - 1;   // clamp: loads stay in-bounds
    const int nc1   = nCol1 < N ? nCol1 : N - 1;

    int aOff = mRow * K + kHi;

    // Two scalar base pointers (K-row ka and ka+1), one voffset per fragment.
    const float* __restrict__ Blo = B;
    const float* __restrict__ Bhi = BT ? B : (B + ldb);
    int b0Off, b1Off, bStep;
    if (BT) { b0Off = nc0 * ldb + kHi; b1Off = nc1 * ldb + kHi; bStep = 4; }
    else    { b0Off = kHi * ldb + nc0; b1Off = kHi * ldb + nc1; bStep = 4 * ldb; }

    v8f c0 = {};
    v8f c1 = {};

    // ---- software pipeline: stage 0 ----
    v2f a = *(const v2f*)(A + aOff);
    v2f b0, b1;
    if (BT) {
        b0 = *(const v2f*)(Blo + b0Off);
        b1 = *(const v2f*)(Blo + b1Off);
    } else {
        b0.x = Blo[b0Off]; b0.y = Bhi[b0Off];
        b1.x = Blo[b1Off]; b1.y = Bhi[b1Off];
    }

    #pragma unroll 2
    for (int k = 0; k < K - 4; k += 4) {
        aOff += 4; b0Off += bStep; b1Off += bStep;

        v2f an = *(const v2f*)(A + aOff);
        v2f b0n, b1n;
        if (BT) {
            b0n = *(const v2f*)(Blo + b0Off);
            b1n = *(const v2f*)(Blo + b1Off);
        } else {
            b0n.x = Blo[b0Off]; b0n.y = Bhi[b0Off];
            b1n.x = Blo[b1Off]; b1n.y = Bhi[b1Off];
        }
        if ((k & 127) == 0) __builtin_prefetch(A + aOff + 256, 0, 1);

        c0 = __builtin_amdgcn_wmma_f32_16x16x4_f32(false, a, false, b0,
                                                   (short)0, c0, false, false);
        c1 = __builtin_amdgcn_wmma_f32_16x16x4_f32(false, a, false, b1,
                                                   (short)0, c1, false, false);
        a = an; b0 = b0n; b1 = b1n;
    }
    c0 = __builtin_amdgcn_wmma_f32_16x16x4_f32(false, a, false, b0,
                                               (short)0, c0, false, false);
    c1 = __builtin_amdgcn_wmma_f32_16x16x4_f32(false, a, false, b1,
                                               (short)0, c1, false, false);

    // Epilogue: C/D layout -> VGPR r holds M = r + 8*(lane>=16), N = lane&15.
    const int mOff  = (lane >> 4) * 8;
    const float bs0 = bias[nc0];
    const float bs1 = bias[nc1];
    #pragma unroll
    for (int r = 0; r < 8; ++r) {
        const int row = mBase + mOff + r;
        float v0 = c0[r] + bs0;
        float v1 = c1[r] + bs1;
        if (RELU) { v0 = v0 > 0.0f ? v0 : 0.0f; v1 = v1 > 0.0f ? v1 : 0.0f; }
        if (nCol0 < N) Dst[row * ldc + nCol0] = v0;
        if (nCol1 < N) Dst[row * ldc + nCol1] = v1;
    }
}

// ---------------------------------------------------------------------------
// Host-side launch: seq -> tmat(+pad) -> 3 WMMA GEMMs, all on `stream`.
// Workspace layout (floats): Cflat[B*KPAD] | Tmat[KPAD*H] | h0[B*H] | h1[B*H]
// Output layout: out[B*O] then relu_masks[B*L] (as 0.0/1.0 floats).
// ---------------------------------------------------------------------------
extern "C" void kernel_launch(void* const* d_in, const int* in_sizes, int n_in,
                              void* d_out, int out_size, void* d_ws, size_t ws_size,
                              hipStream_t stream) {
    const float* x    = (const float*)d_in[0];
    const float* W    = (const float*)d_in[1];
    const float* bv   = (const float*)d_in[2];
    const float* Wb0  = (const float*)d_in[3];
    const float* bb0  = (const float*)d_in[4];
    const float* Wb1  = (const float*)d_in[5];
    const float* bb1  = (const float*)d_in[6];
    const float* Wout = (const float*)d_in[7];
    const float* bout = (const float*)d_in[8];

    float* out   = (float*)d_out;
    float* masks = out + (size_t)BSZ * OO;

    float* ws    = (float*)d_ws;
    float* Cflat = ws;
    float* Tmat  = Cflat + (size_t)BSZ * KPAD;
    float* h0    = Tmat + (size_t)KPAD * HH;
    float* h1    = h0 + (size_t)BSZ * HH;

    seq_kernel<<<BSZ, 64, 0, stream>>>(x, W, bv, Cflat, masks);
    tmat_kernel<<<LL, 256, (size_t)HH * 257 * sizeof(float), stream>>>(W, Wb0, Tmat);
    tpad_kernel<<<KPAD - KAUG, 256, 0, stream>>>(Tmat);

    dim3 gBig(BSZ / 64, HH / 32);
    gemm_wmma<0, 1><<<gBig, 128, 0, stream>>>(Cflat, Tmat, bb0, h0, HH, KPAD, HH, HH);
    gemm_wmma<1, 1><<<gBig, 128, 0, stream>>>(h0, Wb1, bb1, h1, HH, HH, HH, HH);
    dim3 gOut(BSZ / 64, 1);
    gemm_wmma<1, 0><<<gOut, 128, 0, stream>>>(h1, Wout, bout, out, OO, HH, HH, OO);
}